// hapTransformer_65180423684215
// MI455X (gfx1250) — compile-verified
//
#include <hip/hip_runtime.h>
#include <stdint.h>

typedef float v2f __attribute__((ext_vector_type(2)));
typedef float v4f __attribute__((ext_vector_type(4)));
typedef float v8f __attribute__((ext_vector_type(8)));

#define HIDDEN 128
#define NBLK 256
#define TPB 256
#define WPB 8                 // waves per block (wave32)
#define TILES_PER_WAVE 8      // 256 blk * 8 waves * 8 tiles * 16 feat = 262144
#define RSCALE 0.08838834764831845f  // 1/sqrt(128)

// ws layout in floats
#define WS_W    0      // padded B operand: 16 rows x 128 (rows 4..15 zero)
#define WS_Q    2048   // q[128]
#define WS_PART 2176   // NBLK * 516 block partials (acc[4][128] + l[4])
#define PART_STRIDE 516

// ---------------- prep: q = Wq*S + bq ; w_h = Wk_h^T q_h (padded to 16 rows) ---
__global__ void __launch_bounds__(128) prep_kernel(
    const float* __restrict__ S, const float* __restrict__ Wq,
    const float* __restrict__ bq, const float* __restrict__ Wk,
    float* __restrict__ ws) {
  __shared__ float sS[HIDDEN];
  __shared__ float sq[HIDDEN];
  const int t = threadIdx.x;            // 128 threads
  sS[t] = S[t];
  __syncthreads();
  float q = bq[t];
  for (int d = 0; d < HIDDEN; ++d) q += Wq[t * HIDDEN + d] * sS[d];
  sq[t] = q;
  ws[WS_Q + t] = q;
  __syncthreads();
  for (int h = 0; h < 4; ++h) {
    float w = 0.f;
    for (int i = 0; i < 32; ++i)
      w += sq[h * 32 + i] * Wk[(h * 32 + i) * HIDDEN + t];
    ws[WS_W + h * HIDDEN + t] = w;
  }
  for (int r = 4; r < 16; ++r) ws[WS_W + r * HIDDEN + t] = 0.f;  // pad heads
}

// ---------------- main fused pass: scores via WMMA f32, exp, weighted sums ----
__global__ void __launch_bounds__(TPB) attend_kernel(
    const float* __restrict__ X, const float* __restrict__ emb,
    const uint8_t* __restrict__ mask, float* __restrict__ ws) {
  const int tid  = threadIdx.x;
  const int lane = tid & 31;
  const int wave = tid >> 5;
  const int n16  = lane & 15;   // A: M row / B: N col / D: N col
  const int hi   = lane >> 4;   // K half select
  const float* __restrict__ bpt = ws + WS_W + n16 * HIDDEN + 2 * hi;

  v4f acc0 = {0.f,0.f,0.f,0.f}, acc1 = {0.f,0.f,0.f,0.f};
  v4f acc2 = {0.f,0.f,0.f,0.f}, acc3 = {0.f,0.f,0.f,0.f};
  float l0 = 0.f, l1 = 0.f, l2 = 0.f, l3 = 0.f;

  const int g = blockIdx.x * WPB + wave;   // global wave id
  for (int tt = 0; tt < TILES_PER_WAVE; ++tt) {
    const int f0 = (g * TILES_PER_WAVE + tt) * 16;
    // ---- raw scores: D[16f x 16h] += A(emb 16x4) * B(w^T 4x16), K=128 ----
    const float* __restrict__ aptr = emb + (size_t)(f0 + n16) * HIDDEN + 2 * hi;
    v8f c = {};
    #pragma unroll
    for (int kc = 0; kc < 32; ++kc) {
      v2f a = *(const v2f*)(aptr + 4 * kc);
      v2f b = *(const v2f*)(bpt + 4 * kc);
      c = __builtin_amdgcn_wmma_f32_16x16x4_f32(
          false, a, false, b, (short)0, c, false, false);
    }
    // ---- scale by X[f]/sqrt(D) and exponentiate (scores bounded, no max) ----
    const float* xp = X + f0 + 8 * hi;
    v4f xa = *(const v4f*)(xp);
    v4f xb = *(const v4f*)(xp + 4);
    float xv[8] = {xa.x, xa.y, xa.z, xa.w, xb.x, xb.y, xb.z, xb.w};
    float e[8];
    #pragma unroll
    for (int r = 0; r < 8; ++r) e[r] = __expf(c[r] * xv[r] * RSCALE);
    const uint32_t* mwp = (const uint32_t*)(mask + f0);
    uint32_t mword[4] = {mwp[0], mwp[1], mwp[2], mwp[3]};
    // ---- rank-1 accumulation: each lane owns dims 4*lane..4*lane+3 ----
    #pragma unroll
    for (int f = 0; f < 16; ++f) {
      v4f ev = *(const v4f*)(emb + (size_t)(f0 + f) * HIDDEN + 4 * lane);
      float Xf = __shfl(xv[f & 7], (f < 8) ? 0 : 16, 32);
      float mf = (float)((mword[f >> 2] >> ((f & 3) * 8)) & 1u);
      #pragma unroll
      for (int h = 0; h < 4; ++h) {
        float pv = __shfl(e[f & 7], (f < 8) ? h : (h + 16), 32);
        float pm = pv * mf;      // masked-out features contribute nothing
        float tw = pm * Xf;      // p * X  (enc = X*emb)
        if      (h == 0) { l0 += pm; acc0 += tw * ev; }
        else if (h == 1) { l1 += pm; acc1 += tw * ev; }
        else if (h == 2) { l2 += pm; acc2 += tw * ev; }
        else             { l3 += pm; acc3 += tw * ev; }
      }
    }
  }
  // ---- block-level reduction of (acc[4][128], l[4]) across 8 waves ----
  __shared__ float red[WPB][PART_STRIDE];
  float* rw = red[wave];
  #pragma unroll
  for (int i = 0; i < 4; ++i) {
    rw[0 * HIDDEN + 4 * lane + i] = acc0[i];
    rw[1 * HIDDEN + 4 * lane + i] = acc1[i];
    rw[2 * HIDDEN + 4 * lane + i] = acc2[i];
    rw[3 * HIDDEN + 4 * lane + i] = acc3[i];
  }
  if (lane == 0) {
    rw[512 + 0] = l0; rw[512 + 1] = l1; rw[512 + 2] = l2; rw[512 + 3] = l3;
  }
  __syncthreads();
  for (int o = tid; o < PART_STRIDE; o += TPB) {
    float s = 0.f;
    #pragma unroll
    for (int w = 0; w < WPB; ++w) s += red[w][o];
    ws[WS_PART + (size_t)blockIdx.x * PART_STRIDE + o] = s;
  }
}

// ---------------- final: reduce partials + all remaining tiny math -----------
__global__ void __launch_bounds__(256) finish_kernel(
    const float* __restrict__ ws,
    const float* __restrict__ pWv, const float* __restrict__ pbv,
    const float* __restrict__ pWo, const float* __restrict__ pbo,
    const float* __restrict__ sWq, const float* __restrict__ sbq,
    const float* __restrict__ sWv, const float* __restrict__ sbv,
    const float* __restrict__ sWo, const float* __restrict__ sbo,
    const float* __restrict__ Wc,  const float* __restrict__ bc,
    float* __restrict__ out) {
  __shared__ float sS[512];   // acc[4][128]
  __shared__ float sL[4];
  __shared__ float sO[128], sH1[128], sO2[128], sH2[128], lg[2];
  const int t = threadIdx.x;  // 256 threads
  for (int o = t; o < PART_STRIDE; o += 256) {
    float s = 0.f;
    const float* p = ws + WS_PART + o;
    for (int b = 0; b < NBLK; ++b) s += p[(size_t)b * PART_STRIDE];
    if (o < 512) sS[o] = s; else sL[o - 512] = s;
  }
  __syncthreads();
  if (t < 128) {              // attn_h = Wv_h (acc_h / l_h) + bv ; O = q + attn
    const int h = t >> 5;
    float dot = 0.f;
    for (int d = 0; d < HIDDEN; ++d) dot += pWv[t * HIDDEN + d] * sS[h * HIDDEN + d];
    sO[t] = ws[WS_Q + t] + pbv[t] + dot / sL[h];
  }
  __syncthreads();
  if (t < 128) {              // h1 = O + relu(Wo O + bo)
    float r = pbo[t];
    for (int d = 0; d < HIDDEN; ++d) r += pWo[t * HIDDEN + d] * sO[d];
    sH1[t] = sO[t] + fmaxf(r, 0.f);
  }
  __syncthreads();
  if (t < 128) {              // SAB with 1 token: softmax==1 -> O2 = q2 + v2
    float q2 = sbq[t], v2 = sbv[t];
    for (int d = 0; d < HIDDEN; ++d) {
      q2 += sWq[t * HIDDEN + d] * sH1[d];
      v2 += sWv[t * HIDDEN + d] * sH1[d];
    }
    sO2[t] = q2 + v2;
  }
  __syncthreads();
  if (t < 128) {              // h2 = O2 + relu(Wo2 O2 + bo2)
    float r = sbo[t];
    for (int d = 0; d < HIDDEN; ++d) r += sWo[t * HIDDEN + d] * sO2[d];
    sH2[t] = sO2[t] + fmaxf(r, 0.f);
  }
  __syncthreads();
  if (t < 2) {                // logits + softmax
    float v = bc[t];
    for (int d = 0; d < HIDDEN; ++d) v += Wc[t * HIDDEN + d] * sH2[d];
    lg[t] = v;
  }
  __syncthreads();
  if (t < 2) {
    float mx = fmaxf(lg[0], lg[1]);
    float e0 = __expf(lg[0] - mx), e1 = __expf(lg[1] - mx);
    out[t] = ((t == 0) ? e0 : e1) / (e0 + e1);
  }
}

extern "C" void kernel_launch(void* const* d_in, const int* in_sizes, int n_in,
                              void* d_out, int out_size, void* d_ws, size_t ws_size,
                              hipStream_t stream) {
  const float*   X    = (const float*)d_in[0];
  const float*   emb  = (const float*)d_in[1];
  const float*   S    = (const float*)d_in[2];
  const float*   Wc   = (const float*)d_in[3];
  const float*   bc   = (const float*)d_in[4];
  // d_in[5] = tim (unused)
  const uint8_t* mask = (const uint8_t*)d_in[6];
  const float* pWq = (const float*)d_in[7];
  const float* pbq = (const float*)d_in[8];
  const float* pWk = (const float*)d_in[9];
  // d_in[10] = pma_bk: constant shift per head, cancels in softmax
  const float* pWv = (const float*)d_in[11];
  const float* pbv = (const float*)d_in[12];
  const float* pWo = (const float*)d_in[13];
  const float* pbo = (const float*)d_in[14];
  const float* sWq = (const float*)d_in[15];
  const float* sbq = (const float*)d_in[16];
  // d_in[17..18] = sab_Wk/bk: dead (softmax over a single key == 1)
  const float* sWv = (const float*)d_in[19];
  const float* sbv = (const float*)d_in[20];
  const float* sWo = (const float*)d_in[21];
  const float* sbo = (const float*)d_in[22];
  float* ws  = (float*)d_ws;
  float* out = (float*)d_out;

  prep_kernel<<<1, 128, 0, stream>>>(S, pWq, pbq, pWk, ws);
  attend_kernel<<<NBLK, TPB, 0, stream>>>(X, emb, mask, ws);
  finish_kernel<<<1, 256, 0, stream>>>(ws, pWv, pbv, pWo, pbo,
                                       sWq, sbq, sWv, sbv, sWo, sbo,
                                       Wc, bc, out);
}